// RnnEprop1_48430051229763
// MI455X (gfx1250) — compile-verified
//
#include <hip/hip_runtime.h>
#include <hip/hip_bf16.h>
#include <cstddef>

// ---------------- problem constants ----------------
constexpr int T  = 128;
constexpr int Bn = 32;
constexpr int I  = 256;
constexpr int H  = 1024;
constexpr int O  = 128;

// ---------------- workspace layout (floats) ----------------
constexpr size_t EV_WIH = 0;                               // [Bn,H,I]
constexpr size_t EV_WHH = EV_WIH + (size_t)Bn * H * I;     // [Bn,H,H]
constexpr size_t H_BUF  = EV_WHH + (size_t)Bn * H * H;     // 2 x [Bn,H] ping-pong
constexpr size_t COEF   = H_BUF  + (size_t)2 * Bn * H;     // [Bn,H]
constexpr size_t ERRB   = COEF   + (size_t)Bn * H;         // [Bn,O]
constexpr size_t WT_IN  = ERRB   + (size_t)Bn * O;         // [I,H]
constexpr size_t WT_HH  = WT_IN  + (size_t)I * H;          // [H,H]
constexpr size_t WT_OUT = WT_HH  + (size_t)H * H;          // [H,O]

// ---------------- d_out layout (floats) ----------------
constexpr size_t YS_OFF    = 0;                            // [T*Bn,O]
constexpr size_t HF_OFF    = YS_OFF   + (size_t)T * Bn * O;
constexpr size_t GWOUT_OFF = HF_OFF   + (size_t)Bn * H;    // [O,H]
constexpr size_t GWIH_OFF  = GWOUT_OFF + (size_t)O * H;    // [H,I]
constexpr size_t GWHH_OFF  = GWIH_OFF  + (size_t)H * I;    // [H,H]

typedef float v2f __attribute__((ext_vector_type(2)));
typedef float v8f __attribute__((ext_vector_type(8)));
typedef float f4  __attribute__((ext_vector_type(4)));

// ---------------------------------------------------------------------------
// fp32 WMMA 16x16 tile GEMM accumulate: C += A[rowBase:rowBase+16, 0:K] *
// Bt[0:K, colBase:colBase+16], where A is row-major [M,lda] and Bt is
// row-major [K,ldb] (i.e. Bt[k][n]).  Fragment layouts per CDNA5 ISA 7.12.2:
//   A (16x4 f32): lanes 0-15 M=0-15; v0 holds K={0|2}, v1 holds K={1|3}
//   B (4x16 f32): row striped across lanes; v0 -> K={0|1}, v1 -> K={2|3}
//   C/D (16x16): vgpr g, lanes 0-15 -> M=g, lanes 16-31 -> M=g+8
// ---------------------------------------------------------------------------
__device__ inline v8f wmma_gemm_acc(v8f c,
                                    const float* __restrict__ A, int lda, int rowBase,
                                    const float* __restrict__ Bt, int ldb, int colBase,
                                    int K, int lane) {
  const int ml = lane & 15;
  const int hi = lane >> 4;
  const float* arow = A + (size_t)(rowBase + ml) * lda;
  const float* bcol = Bt + colBase + ml;
  for (int k0 = 0; k0 < K; k0 += 4) {
    v2f a, b;
    a.x = arow[k0 + 2 * hi];
    a.y = arow[k0 + 2 * hi + 1];
    b.x = bcol[(size_t)(k0 + hi) * ldb];
    b.y = bcol[(size_t)(k0 + 2 + hi) * ldb];
    c = __builtin_amdgcn_wmma_f32_16x16x4_f32(false, a, false, b, (short)0, c,
                                              false, false);
  }
  return c;
}

// ---------------------------------------------------------------------------
// K1a: h_new = tanh(x_t @ W_in^T + h_old @ W_hh^T)   [Bn,H]
// grid <<<16,256>>> : 128 waves, 2 M-tiles x 64 N-tiles
// ---------------------------------------------------------------------------
__global__ void k_hnew(const float* __restrict__ x_t, const float* __restrict__ h_old,
                       const float* __restrict__ Wt_in, const float* __restrict__ Wt_hh,
                       float* __restrict__ h_new, float* __restrict__ hf, int write_hf) {
  const int wave = (blockIdx.x * blockDim.x + threadIdx.x) >> 5;
  const int lane = threadIdx.x & 31;
  const int mt = wave & 1;
  const int nt = wave >> 1;
  v8f c = {0.f, 0.f, 0.f, 0.f, 0.f, 0.f, 0.f, 0.f};
  c = wmma_gemm_acc(c, x_t,   I, mt * 16, Wt_in, H, nt * 16, I, lane);
  c = wmma_gemm_acc(c, h_old, H, mt * 16, Wt_hh, H, nt * 16, H, lane);
  const int ml = lane & 15, hi = lane >> 4;
  const int col = nt * 16 + ml;
  for (int g = 0; g < 8; ++g) {
    const int row = mt * 16 + g + 8 * hi;
    const float v = tanhf(c[g]);
    h_new[row * H + col] = v;
    if (write_hf) hf[row * H + col] = v;
  }
}

// ---------------------------------------------------------------------------
// K1b: o = h_new @ W_out^T [Bn,O]; ys_t = o; err = o - target_t
// grid <<<1,512>>> : 16 waves, 2 M-tiles x 8 N-tiles, K=H
// ---------------------------------------------------------------------------
__global__ void k_out(const float* __restrict__ h_new, const float* __restrict__ Wt_out,
                      const float* __restrict__ tgt_t, float* __restrict__ ys_t,
                      float* __restrict__ err) {
  const int wave = threadIdx.x >> 5;
  const int lane = threadIdx.x & 31;
  const int mt = wave & 1;
  const int nt = wave >> 1;
  v8f c = {0.f, 0.f, 0.f, 0.f, 0.f, 0.f, 0.f, 0.f};
  c = wmma_gemm_acc(c, h_new, H, mt * 16, Wt_out, O, nt * 16, H, lane);
  const int ml = lane & 15, hi = lane >> 4;
  const int col = nt * 16 + ml;
  for (int g = 0; g < 8; ++g) {
    const int row = mt * 16 + g + 8 * hi;
    const float o = c[g];
    ys_t[row * O + col] = o;
    err[row * O + col]  = o - tgt_t[row * O + col];
  }
}

// ---------------------------------------------------------------------------
// K1c: L = err @ W_out  [Bn,H];  coef = L * (1 - h_new^2)
// W_out is [O,H] row-major -> already B-operand layout Bt[k=o][n=j].
// grid <<<16,256>>>
// ---------------------------------------------------------------------------
__global__ void k_learn(const float* __restrict__ err, const float* __restrict__ W_out,
                        const float* __restrict__ h_new, float* __restrict__ coef) {
  const int wave = (blockIdx.x * blockDim.x + threadIdx.x) >> 5;
  const int lane = threadIdx.x & 31;
  const int mt = wave & 1;
  const int nt = wave >> 1;
  v8f c = {0.f, 0.f, 0.f, 0.f, 0.f, 0.f, 0.f, 0.f};
  c = wmma_gemm_acc(c, err, O, mt * 16, W_out, H, nt * 16, O, lane);
  const int ml = lane & 15, hi = lane >> 4;
  const int col = nt * 16 + ml;
  for (int g = 0; g < 8; ++g) {
    const int row = mt * 16 + g + 8 * hi;
    const float hn = h_new[row * H + col];
    coef[row * H + col] = c[g] * (1.f - hn * hn);
  }
}

// ---------------------------------------------------------------------------
// K1d: g_wout_mean[o,hh] += (1/Bn) * sum_b err[b,o]*h_new[b,hh]
// ---------------------------------------------------------------------------
__global__ void k_gwout(const float* __restrict__ err, const float* __restrict__ h_new,
                        float* __restrict__ g) {
  const int idx = blockIdx.x * blockDim.x + threadIdx.x;  // O*H threads
  const int o  = idx >> 10;        // / H
  const int hh = idx & (H - 1);    // % H
  float acc = 0.f;
  for (int b = 0; b < Bn; ++b) acc += err[b * O + o] * h_new[b * H + hh];
  g[idx] += acc * (1.f / Bn);
}

// ---------------------------------------------------------------------------
// K2: fused trace update + mean-gradient accumulation (the bandwidth hog).
// Single pass over ev:  e = D[b,j]*e + inp[b,k];  g[j,k] += sum_b c2[b,j]*e
// Batch loop lives in-thread -> no atomics, g RMW once per (j,k) per step.
// whh: block = 256 threads (float4 over k=0..1023), grid = H (one block per j)
// ---------------------------------------------------------------------------
__global__ void k_trace_whh(const float* __restrict__ h_old, const float* __restrict__ coef,
                            const float* __restrict__ W_hh, float* __restrict__ ev,
                            float* __restrict__ g) {
  const int j = blockIdx.x;
  const int k = threadIdx.x * 4;
  __shared__ float Ds[Bn], Cs[Bn];
  if (threadIdx.x < Bn) {
    const float hj = h_old[threadIdx.x * H + j];
    Ds[threadIdx.x] = W_hh[(size_t)j * H + j] * (1.f - hj * hj);
    Cs[threadIdx.x] = coef[threadIdx.x * H + j] * (1e-5f / Bn);
  }
  __syncthreads();
  f4 acc = {0.f, 0.f, 0.f, 0.f};
  float* evj = ev + (size_t)j * H + k;
  for (int b = 0; b < Bn; ++b) {
    f4 e  = *reinterpret_cast<const f4*>(evj + (size_t)b * H * H);
    f4 hv = *reinterpret_cast<const f4*>(h_old + b * H + k);
    e = Ds[b] * e + hv;
    *reinterpret_cast<f4*>(evj + (size_t)b * H * H) = e;
    acc += Cs[b] * e;
  }
  f4* gp = reinterpret_cast<f4*>(g + (size_t)j * H + k);
  *gp = *gp + acc;
}

// wih: block = 64 threads (float4 over k=0..255), grid = H
__global__ void k_trace_wih(const float* __restrict__ h_old, const float* __restrict__ coef,
                            const float* __restrict__ W_hh, const float* __restrict__ x_t,
                            float* __restrict__ ev, float* __restrict__ g) {
  const int j = blockIdx.x;
  const int k = threadIdx.x * 4;
  __shared__ float Ds[Bn], Cs[Bn];
  if (threadIdx.x < Bn) {
    const float hj = h_old[threadIdx.x * H + j];
    Ds[threadIdx.x] = W_hh[(size_t)j * H + j] * (1.f - hj * hj);
    Cs[threadIdx.x] = coef[threadIdx.x * H + j] * (1e-3f / Bn);
  }
  __syncthreads();
  f4 acc = {0.f, 0.f, 0.f, 0.f};
  float* evj = ev + (size_t)j * I + k;
  for (int b = 0; b < Bn; ++b) {
    f4 e  = *reinterpret_cast<const f4*>(evj + (size_t)b * H * I);
    f4 xv = *reinterpret_cast<const f4*>(x_t + b * I + k);
    e = Ds[b] * e + xv;
    *reinterpret_cast<f4*>(evj + (size_t)b * H * I) = e;
    acc += Cs[b] * e;
  }
  f4* gp = reinterpret_cast<f4*>(g + (size_t)j * I + k);
  *gp = *gp + acc;
}

// one-shot transpose: src [R,C] row-major -> dst [C,R]
__global__ void k_transpose(const float* __restrict__ src, float* __restrict__ dst,
                            int R, int C) {
  const int idx = blockIdx.x * blockDim.x + threadIdx.x;
  if (idx >= R * C) return;
  const int r = idx / C, cc = idx % C;
  dst[(size_t)cc * R + r] = src[(size_t)r * C + cc];
}

// ---------------------------------------------------------------------------
extern "C" void kernel_launch(void* const* d_in, const int* in_sizes, int n_in,
                              void* d_out, int out_size, void* d_ws, size_t ws_size,
                              hipStream_t stream) {
  (void)in_sizes; (void)n_in; (void)out_size; (void)ws_size;
  const float* x      = (const float*)d_in[0];  // [T,Bn,I]
  const float* target = (const float*)d_in[1];  // [T,Bn,O]
  const float* h0     = (const float*)d_in[2];  // [Bn,H]
  const float* W_in   = (const float*)d_in[3];  // [H,I]
  const float* W_hh   = (const float*)d_in[4];  // [H,H]
  const float* W_out  = (const float*)d_in[5];  // [O,H]
  float* out = (float*)d_out;
  float* w   = (float*)d_ws;

  float* ev_wih = w + EV_WIH;
  float* ev_whh = w + EV_WHH;
  float* h_buf  = w + H_BUF;
  float* coefb  = w + COEF;
  float* errb   = w + ERRB;
  float* Wt_in  = w + WT_IN;
  float* Wt_hh  = w + WT_HH;
  float* Wt_out = w + WT_OUT;

  // init: zero traces + mean-gradient accumulators (which live in d_out)
  hipMemsetAsync(ev_wih, 0, (size_t)(Bn * H * I + Bn * H * H) * sizeof(float), stream);
  hipMemsetAsync(out + GWOUT_OFF, 0,
                 (size_t)(O * H + H * I + H * H) * sizeof(float), stream);
  hipMemcpyAsync(h_buf, h0, (size_t)Bn * H * sizeof(float),
                 hipMemcpyDeviceToDevice, stream);

  // one-time weight transposes for coalesced WMMA B-operand loads
  k_transpose<<<(H * I + 255) / 256, 256, 0, stream>>>(W_in,  Wt_in,  H, I);
  k_transpose<<<(H * H + 255) / 256, 256, 0, stream>>>(W_hh,  Wt_hh,  H, H);
  k_transpose<<<(O * H + 255) / 256, 256, 0, stream>>>(W_out, Wt_out, O, H);

  for (int t = 0; t < T; ++t) {
    const float* x_t = x + (size_t)t * Bn * I;
    const float* z_t = target + (size_t)t * Bn * O;
    float* ho = h_buf + (size_t)(t & 1) * Bn * H;
    float* hn = h_buf + (size_t)((t + 1) & 1) * Bn * H;

    k_hnew<<<16, 256, 0, stream>>>(x_t, ho, Wt_in, Wt_hh, hn,
                                   out + HF_OFF, (t == T - 1) ? 1 : 0);
    k_out<<<1, 512, 0, stream>>>(hn, Wt_out, z_t,
                                 out + YS_OFF + (size_t)t * Bn * O, errb);
    k_learn<<<16, 256, 0, stream>>>(errb, W_out, hn, coefb);
    k_gwout<<<(O * H) / 256, 256, 0, stream>>>(errb, hn, out + GWOUT_OFF);
    k_trace_wih<<<H, I / 4, 0, stream>>>(ho, coefb, W_hh, x_t, ev_wih,
                                         out + GWIH_OFF);
    k_trace_whh<<<H, H / 4, 0, stream>>>(ho, coefb, W_hh, ev_whh,
                                         out + GWHH_OFF);
  }
}